// DenoisedSignalFeatureExtractor_32315333935261
// MI455X (gfx1250) — compile-verified
//
#include <hip/hip_runtime.h>
#include <math.h>

// ---------------------------------------------------------------------------
// Mamba signal feature extractor for MI455X (gfx1250, wave32).
// Dense contractions (in_proj / x_proj / dt / out_proj / te-conv / ae-conv)
// run on V_WMMA_F32_16X16X4_F32; convs are lowered to GEMM via implicit
// im2col on zero-padded (B, L+2, C) activation buffers. Each wave computes
// 4 M-tiles per N-tile so one B-fragment load feeds 4 WMMA issues; the
// k-loop is software-pipelined (next-step fragments prefetched into regs
// before the current step's WMMAs issue) so loads overlap matrix ops.
// Selective scan: 1 block per batch, 256 threads = 256 channels, state in
// registers, B/C vectors broadcast through LDS.
// ---------------------------------------------------------------------------

#define BB   32
#define LL   1024
#define BLM  (BB * LL)        // 32768 rows, multiple of 64
#define DM   128
#define DI   256
#define PSTR (LL + 2)         // padded row stride per batch

typedef __attribute__((ext_vector_type(2))) float v2f;
typedef __attribute__((ext_vector_type(8))) float v8f;

__device__ __forceinline__ float geluf(float x) {
  return 0.5f * x * (1.f + erff(x * 0.70710678118654752f));
}
__device__ __forceinline__ float siluf(float x) {
  return x / (1.f + __expf(-x));
}
__device__ __forceinline__ float softplusf(float x) {
  if (x > 20.f) return x;
  if (x < -20.f) return __expf(x);
  return log1pf(__expf(x));
}

// ------------------------- generic WMMA fp32 GEMM ---------------------------
// Y[M,N] = act( X[M,K] * W[N,K]^T + bias ); wave computes MT=4 16x16 M-tiles.
// NG: N not multiple of 16 -> clamp+select (branch-free, no exec juggling).
// PADROWS: X rows live in a per-batch zero-padded (rpb+2) buffer (im2col).
// KB: K runs in blocks of 32 at stride `kstride` in X (grouped conv im2col).
// gct/gcs: group column offset = (tn / gct) * gcs added to X columns.
template <int ACT, bool BIAS, bool PADROWS, bool NG, bool KB, int MT>
__global__ void k_gemm(const float* __restrict__ X, const float* __restrict__ W,
                       const float* __restrict__ bias, float* __restrict__ Y,
                       int N, int K, int sx, int sw, int syy,
                       int ntn, long ntiles, int rpb, int pstride,
                       int kstride, int gct, int gcs) {
  long wv = (long)blockIdx.x * 8 + (threadIdx.x >> 5);
  if (wv >= ntiles) return;          // wave-uniform exit (EXEC all-1 for WMMA)
  int lane = threadIdx.x & 31;
  int mg = (int)(wv / ntn), tn = (int)(wv % ntn);
  int hi = lane >> 4;
  int ml = lane & 15;
  int nB = tn * 16 + ml;
  bool nok = NG ? (nB < N) : true;
  int nBc = NG ? (nok ? nB : N - 1) : nB;  // clamp: always-legal address
  int coladd = (gct > 0) ? (tn / gct) * gcs : 0;
  long xoff[MT];
#pragma unroll
  for (int t = 0; t < MT; ++t) {
    int mA = (mg * MT + t) * 16 + ml;
    if (PADROWS) {
      int bb = mA / rpb;
      int ll = mA - bb * rpb;
      xoff[t] = ((long)bb * pstride + ll) * (long)sx + coladd;
    } else {
      xoff[t] = (long)mA * (long)sx + coladd;
    }
  }
  v8f acc[MT];
#pragma unroll
  for (int t = 0; t < MT; ++t) acc[t] = {0.f, 0.f, 0.f, 0.f, 0.f, 0.f, 0.f, 0.f};

  const int nkb = KB ? (K >> 5) : 1;     // K-blocks (32-wide for grouped conv)
  const int kbsz = KB ? 32 : K;
  const float* wp = W + (long)nBc * (long)sw + 2 * hi;
  for (int kb = 0; kb < nkb; ++kb) {
    const float* xp[MT];
#pragma unroll
    for (int t = 0; t < MT; ++t)
      xp[t] = X + xoff[t] + (KB ? (long)kb * kstride : 0) + 2 * hi;

    // software pipeline: prefetch step k0+4 while WMMAs consume step k0
    v2f cb, ca[MT];
    cb.x = wp[0]; cb.y = wp[1];
    if (NG && !nok) { cb.x = 0.f; cb.y = 0.f; }
#pragma unroll
    for (int t = 0; t < MT; ++t) { ca[t].x = xp[t][0]; ca[t].y = xp[t][1]; }
#pragma unroll 2
    for (int k0 = 4; k0 < kbsz; k0 += 4) {
      v2f nb, na[MT];
      nb.x = wp[k0]; nb.y = wp[k0 + 1];
      if (NG && !nok) { nb.x = 0.f; nb.y = 0.f; }
#pragma unroll
      for (int t = 0; t < MT; ++t) { na[t].x = xp[t][k0]; na[t].y = xp[t][k0 + 1]; }
#pragma unroll
      for (int t = 0; t < MT; ++t)
        acc[t] = __builtin_amdgcn_wmma_f32_16x16x4_f32(false, ca[t], false, cb,
                                                       (short)0, acc[t], false, false);
      cb = nb;
#pragma unroll
      for (int t = 0; t < MT; ++t) ca[t] = na[t];
    }
#pragma unroll
    for (int t = 0; t < MT; ++t)
      acc[t] = __builtin_amdgcn_wmma_f32_16x16x4_f32(false, ca[t], false, cb,
                                                     (short)0, acc[t], false, false);
    wp += kbsz;                        // weights stay contiguous across blocks
  }

  if (!NG || nok) {
    float bval = BIAS ? bias[nB] : 0.f;
#pragma unroll
    for (int t = 0; t < MT; ++t) {
      int mbase = (mg * MT + t) * 16 + 8 * hi;
#pragma unroll
      for (int r = 0; r < 8; ++r) {
        float v = acc[t][r] + bval;
        if (ACT == 1) v = softplusf(v);
        Y[(long)(mbase + r) * syy + nB] = v;
      }
    }
  }
}

// ------------- initial conv(32->128,k=7,pad3) + BN + gelu + pos -------------
__global__ void k_proj(const float* __restrict__ x, const float* __restrict__ w,
                       const float* __restrict__ bias, const float* __restrict__ bng,
                       const float* __restrict__ bnb, const float* __restrict__ pos,
                       float* __restrict__ tf) {
  int blk = blockIdx.x;            // 32 * 8 blocks
  int b = blk >> 3;
  int l0 = (blk & 7) * 128;
  __shared__ float xs[32 * 134];   // 32 cin x (128 + 6) halo
  for (int i = threadIdx.x; i < 32 * 134; i += 256) {
    int ci = i / 134, p = i - ci * 134;
    int gl = l0 + p - 3;
    xs[i] = (gl >= 0 && gl < LL) ? x[(long)b * 32 * LL + (long)ci * LL + gl] : 0.f;
  }
  __syncthreads();
  for (int it = 0; it < 64; ++it) {
    int oi = it * 256 + threadIdx.x;
    int c = oi & 127, li = oi >> 7;
    float acc = bias[c];
    const float* wr = w + c * 224;
#pragma unroll
    for (int ci = 0; ci < 32; ++ci) {
      const float* xr = xs + ci * 134 + li;
#pragma unroll
      for (int k = 0; k < 7; ++k) acc += wr[ci * 7 + k] * xr[k];
    }
    float v = geluf(bng[c] * acc + bnb[c]);
    int l = l0 + li;
    // linear interp of pos (128 -> 1024), align_corners=False
    float src = (l + 0.5f) * 0.125f - 0.5f;
    src = fminf(fmaxf(src, 0.f), 127.f);
    int i0 = (int)floorf(src);
    int i1 = min(i0 + 1, 127);
    float fw = src - (float)i0;
    float pe = pos[i0 * DM + c] * (1.f - fw) + pos[i1 * DM + c] * fw;
    tf[((long)b * LL + l) * DM + c] = v + pe;
  }
}

// -------------------- depthwise causal conv k=4 + silu ----------------------
__global__ void k_dwconv(const float* __restrict__ xz, const float* __restrict__ cw,
                         const float* __restrict__ cb, float* __restrict__ u) {
  long idx = (long)blockIdx.x * 256 + threadIdx.x;   // BL * 256
  int c = (int)(idx & 255);
  long bl = idx >> 8;
  int l = (int)(bl & 1023);
  long b = bl >> 10;
  float acc = cb[c];
#pragma unroll
  for (int k = 0; k < 4; ++k) {
    int ll = l - 3 + k;
    if (ll >= 0) acc += cw[c * 4 + k] * xz[(((long)b << 10) + ll) * 512 + c];
  }
  u[idx] = siluf(acc);
}

// ------------------------------ selective scan ------------------------------
__global__ void k_scan(const float* __restrict__ u, const float* __restrict__ dt,
                       const float* __restrict__ xdbl, const float* __restrict__ xz,
                       const float* __restrict__ A_log, const float* __restrict__ Dp,
                       float* __restrict__ out) {
  int b = blockIdx.x;
  int c = threadIdx.x;               // 256 channels
  float A[16], h[16];
#pragma unroll
  for (int n = 0; n < 16; ++n) { A[n] = -__expf(A_log[c * 16 + n]); h[n] = 0.f; }
  float Dc = Dp[c];
  __shared__ float BC[32];           // Bm[0..15], Cm[0..15]
  for (int l = 0; l < LL; ++l) {
    long row = ((long)b << 10) + l;
    if (c < 32) {
      int n = c & 15;
      BC[c] = xdbl[row * 40 + (c < 16 ? 8 + n : 24 + n)];
    }
    __syncthreads();
    float ut = u[row * 256 + c];
    float dtt = dt[row * 256 + c];
    float du = dtt * ut;
    float y = 0.f;
#pragma unroll
    for (int n = 0; n < 16; ++n) {
      float dA = __expf(dtt * A[n]);
      h[n] = dA * h[n] + du * BC[n];
      y = fmaf(h[n], BC[16 + n], y);
    }
    float z = xz[row * 512 + 256 + c];
    out[row * 256 + c] = (y + ut * Dc) * siluf(z);
    __syncthreads();
  }
}

// ------- two chained LayerNorms per row; writes (B, L+2, 128) padded --------
__global__ void k_ln(const float* __restrict__ r, const float* __restrict__ ym,
                     const float* __restrict__ g1, const float* __restrict__ b1,
                     const float* __restrict__ g2, const float* __restrict__ b2,
                     float* __restrict__ outp) {
  long row = blockIdx.x;
  int c = threadIdx.x;               // 128
  __shared__ float red[128];
  __shared__ float stat;
  float y = ym[row * 128 + c];
  red[c] = y; __syncthreads();
  for (int s = 64; s > 0; s >>= 1) { if (c < s) red[c] += red[c + s]; __syncthreads(); }
  if (c == 0) stat = red[0] * (1.f / 128.f);
  __syncthreads();
  float m = stat; float d = y - m;
  __syncthreads();
  red[c] = d * d; __syncthreads();
  for (int s = 64; s > 0; s >>= 1) { if (c < s) red[c] += red[c + s]; __syncthreads(); }
  if (c == 0) stat = red[0] * (1.f / 128.f);
  __syncthreads();
  float y1 = d * rsqrtf(stat + 1e-5f) * g1[c] + b1[c];
  float sv = r[row * 128 + c] + y1;
  __syncthreads();
  red[c] = sv; __syncthreads();
  for (int s = 64; s > 0; s >>= 1) { if (c < s) red[c] += red[c + s]; __syncthreads(); }
  if (c == 0) stat = red[0] * (1.f / 128.f);
  __syncthreads();
  float m2 = stat; float d2 = sv - m2;
  __syncthreads();
  red[c] = d2 * d2; __syncthreads();
  for (int s = 64; s > 0; s >>= 1) { if (c < s) red[c] += red[c + s]; __syncthreads(); }
  if (c == 0) stat = red[0] * (1.f / 128.f);
  __syncthreads();
  int b = (int)(row >> 10), l = (int)(row & 1023);
  outp[((long)b * PSTR + 1 + l) * 128 + c] =
      d2 * rsqrtf(stat + 1e-5f) * g2[c] + b2[c];
}

// te epilogue: out = x + gelu(bn(conv)); x read from padded buffer
__global__ void k_te_epi(const float* __restrict__ conv, const float* __restrict__ xp,
                         const float* __restrict__ bng, const float* __restrict__ bnb,
                         float* __restrict__ out) {
  long idx = (long)blockIdx.x * 256 + threadIdx.x;   // BL * 128
  int c = (int)(idx & 127);
  long bl = idx >> 7;
  int l = (int)(bl & 1023);
  int b = (int)(bl >> 10);
  float cv = conv[idx];
  float xv = xp[((long)b * PSTR + 1 + l) * 128 + c];
  out[idx] = xv + geluf(bng[c] * cv + bnb[c]);
}

// repack te weights (128, 32, 3) -> (128, kk*32+ci)
__global__ void k_wpack_te(const float* __restrict__ tw, float* __restrict__ wp) {
  int i = blockIdx.x * 256 + threadIdx.x;
  if (i >= 128 * 96) return;
  int co = i / 96;
  int k = i - co * 96;
  int kk = k >> 5, ci = k & 31;
  wp[i] = tw[co * 96 + ci * 3 + kk];
}

// --------------- channel attention: pooled MLP -> sigmoid gate --------------
__global__ void k_ca_attn(const float* __restrict__ tf, const float* __restrict__ w1,
                          const float* __restrict__ w2, float* __restrict__ a) {
  int b = blockIdx.x;
  int t = threadIdx.x;               // 256
  int c = t & 127, hh = t >> 7;
  float s = 0.f, m = -3.4e38f;
  for (int l = hh; l < LL; l += 2) {
    float v = tf[(((long)b << 10) + l) * 128 + c];
    s += v; m = fmaxf(m, v);
  }
  __shared__ float ssum[256], smax[256];
  __shared__ float avg[128], mxv[128], hid[32];
  ssum[t] = s; smax[t] = m; __syncthreads();
  if (hh == 0) {
    avg[c] = (ssum[c] + ssum[c + 128]) * (1.f / LL);
    mxv[c] = fmaxf(smax[c], smax[c + 128]);
  }
  __syncthreads();
  if (t < 32) {
    int j = t & 15;
    const float* src = (t < 16) ? avg : mxv;
    float acc = 0.f;
    for (int k = 0; k < 128; ++k) acc += w1[j * 128 + k] * src[k];
    hid[t] = geluf(acc);
  }
  __syncthreads();
  if (t < 128) {
    float acc = 0.f;
#pragma unroll
    for (int j = 0; j < 16; ++j) acc += w2[t * 16 + j] * (hid[j] + hid[16 + j]);
    a[b * 128 + t] = 1.f / (1.f + __expf(-acc));
  }
}

// zero pad rows (one row before/after each batch) of a (B, L+2, 128) buffer
__global__ void k_padzero(float* __restrict__ p) {
  int b = blockIdx.x;
  int t = threadIdx.x;               // 256
  if (t < 128) p[((long)b * PSTR) * 128 + t] = 0.f;
  else         p[((long)b * PSTR + PSTR - 1) * 128 + (t - 128)] = 0.f;
}

// gate + grouped enh conv (groups=32,k=3) -> padded att buffer (B, L+2, 128)
__global__ void k_ca_enh(const float* __restrict__ tf, const float* __restrict__ a,
                         const float* __restrict__ ew, const float* __restrict__ eb,
                         const float* __restrict__ bng, const float* __restrict__ bnb,
                         float* __restrict__ attp) {
  long idx = (long)blockIdx.x * 256 + threadIdx.x;   // BL * 128
  int c = (int)(idx & 127);
  long bl = idx >> 7;
  int l = (int)(bl & 1023);
  int b = (int)(bl >> 10);
  int g = c >> 2;
  const float* ar = a + b * 128;
  float acc = eb[c];
  for (int k = 0; k < 3; ++k) {
    int ll = l - 1 + k;
    if (ll < 0 || ll >= LL) continue;
    const float* xr = tf + (((long)b << 10) + ll) * 128 + g * 4;
#pragma unroll
    for (int ci = 0; ci < 4; ++ci) acc += ew[c * 12 + ci * 3 + k] * xr[ci] * ar[g * 4 + ci];
  }
  float xa = tf[idx] * ar[c];
  attp[((long)b * PSTR + 1 + l) * 128 + c] = xa + 0.1f * geluf(bng[c] * acc + bnb[c]);
}

// repack ae conv weights (128,128,3) -> (128, kk*128+ci)
__global__ void k_wpack_ae(const float* __restrict__ aw, float* __restrict__ wp) {
  int i = blockIdx.x * 256 + threadIdx.x;
  if (i >= 128 * 384) return;
  int co = i / 384;
  int k = i - co * 384;
  int kk = k >> 7, ci = k & 127;
  wp[i] = aw[co * 384 + ci * 3 + kk];
}

// bn + gelu + residual, write enh in (B, C, L) output layout
__global__ void k_ae_epi(const float* __restrict__ conv, const float* __restrict__ attp,
                         const float* __restrict__ bng, const float* __restrict__ bnb,
                         float* __restrict__ enh) {
  long idx = (long)blockIdx.x * 256 + threadIdx.x;   // b*128*1024, l fastest
  int l = (int)(idx & 1023);
  long r = idx >> 10;
  int c = (int)(r & 127);
  int b = (int)(r >> 7);
  long row = ((long)b << 10) + l;
  float cv = conv[row * 128 + c];
  float av = attp[((long)b * PSTR + 1 + l) * 128 + c];
  enh[idx] = geluf(bng[c] * cv + bnb[c]) + av;
}

// mean / max / std pooling over L per (b,c)
__global__ void k_pool(const float* __restrict__ enh, float* __restrict__ pooled,
                       float* __restrict__ gf) {
  int bc = blockIdx.x;
  int b = bc >> 7, c = bc & 127;
  const float* p = enh + (long)bc * LL;
  int t = threadIdx.x;               // 256
  float s = 0.f, s2 = 0.f, m = -3.4e38f;
  for (int l = t; l < LL; l += 256) {
    float v = p[l];
    s += v; s2 += v * v; m = fmaxf(m, v);
  }
  __shared__ float rs[256], rq[256], rm[256];
  rs[t] = s; rq[t] = s2; rm[t] = m; __syncthreads();
  for (int st = 128; st > 0; st >>= 1) {
    if (t < st) { rs[t] += rs[t + st]; rq[t] += rq[t + st]; rm[t] = fmaxf(rm[t], rm[t + st]); }
    __syncthreads();
  }
  if (t == 0) {
    float mean = rs[0] * (1.f / LL);
    float var = rq[0] * (1.f / LL) - mean * mean;
    pooled[b * 384 + c] = mean;
    pooled[b * 384 + 128 + c] = rm[0];
    pooled[b * 384 + 256 + c] = sqrtf(fmaxf(var, 0.f) + 1e-8f);
    gf[b * 128 + c] = mean;
  }
}

// ---------------------------------------------------------------------------
extern "C" void kernel_launch(void* const* d_in, const int* in_sizes, int n_in,
                              void* d_out, int out_size, void* d_ws, size_t ws_size,
                              hipStream_t stream) {
  (void)n_in; (void)out_size; (void)ws_size;
  // JAX pytree flatten sorts dict keys -> 49 param leaves. Detect whether x
  // (unique size 32*32*1024) is first or last.
  const int NP = 49;
  bool x_first = (in_sizes[0] == BB * 32 * LL);
  const float* x = (const float*)d_in[x_first ? 0 : NP];
  const float* P[NP];
  int base = x_first ? 1 : 0;
  for (int i = 0; i < NP; ++i) P[i] = (const float*)d_in[base + i];

  const float *ae_b = P[0], *ae_bnb = P[1], *ae_bng = P[2], *ae_w = P[3];
  const float *ca_bnb = P[38], *ca_bng = P[39], *ca_enhb = P[40], *ca_enhw = P[41];
  const float *ca_w1 = P[42], *ca_w2 = P[43];
  const float *pos = P[44];
  const float *pj_b = P[45], *pj_bnb = P[46], *pj_bng = P[47], *pj_w = P[48];

  float* ws = (float*)d_ws;
  size_t off = 0;
  auto alloc = [&](size_t n) { float* p = ws + off; off += n; return p; };
  float* tfA  = alloc((size_t)BLM * DM);
  float* tfB  = alloc((size_t)BLM * DM);
  float* tfCp = alloc((size_t)BB * PSTR * DM);   // padded LN output (im2col src)
  float* xz   = alloc((size_t)BLM * 512);
  float* u    = alloc((size_t)BLM * DI);
  float* xdbl = alloc((size_t)BLM * 40);
  float* dt   = alloc((size_t)BLM * DI);
  float* sy   = alloc((size_t)BLM * DI);
  float* ym   = alloc((size_t)BLM * DM);
  float* wpte = alloc((size_t)128 * 96);
  float* wpae = alloc((size_t)128 * 384);
  // buffers reused after the mamba layers finish:
  float* attp    = xz;   // 32*1026*128 < BL*512
  float* a_attn  = dt;   // 32*128
  float* convraw = ym;   // BL*128

  float* out = (float*)d_out;
  float* pooled = out;                     // (32, 384)
  float* gf = out + BB * 384;              // (32, 128)
  float* enh = out + BB * 384 + BB * DM;   // (32, 128, 1024)

  const long MT4 = BLM / 16 / 4;           // 512 m-groups

  // stem: conv + BN + gelu + interp pos
  k_proj<<<BB * 8, 256, 0, stream>>>(x, pj_w, pj_b, pj_bng, pj_bnb, pos, tfA);
  k_padzero<<<BB, 256, 0, stream>>>(tfCp);

  float* tin = tfA;
  float* tout = tfB;
  for (int layer = 0; layer < 2; ++layer) {
    const float* const* Q = P + 4 + layer * 17;
    const float *A_log = Q[0], *Dp = Q[1], *conv_b = Q[2], *conv_w = Q[3];
    const float *dt_b = Q[4], *dt_w = Q[5], *in_proj = Q[6];
    const float *ln1b = Q[7], *ln1g = Q[8], *ln2b = Q[9], *ln2g = Q[10];
    const float *out_proj = Q[11], *te_b = Q[12], *te_bnb = Q[13];
    const float *te_bng = Q[14], *te_w = Q[15], *x_proj = Q[16];

    { // in_proj: (BL,512) = tin(BL,128) x W(512,128)^T
      long tiles = MT4 * (512 / 16);
      k_gemm<0, false, false, false, false, 4><<<(int)((tiles + 7) / 8), 256, 0, stream>>>(
          tin, in_proj, nullptr, xz, 512, 128, 128, 128, 512, 512 / 16, tiles,
          0, 0, 0, 0, 0);
    }
    k_dwconv<<<BLM, 256, 0, stream>>>(xz, conv_w, conv_b, u);
    { // x_proj: (BL,40) = u(BL,256) x W(40,256)^T  (N-guarded)
      long tiles = MT4 * 3;
      k_gemm<0, false, false, true, false, 4><<<(int)((tiles + 7) / 8), 256, 0, stream>>>(
          u, x_proj, nullptr, xdbl, 40, 256, 256, 256, 40, 3, tiles,
          0, 0, 0, 0, 0);
    }
    { // dt: softplus(xdbl[:, :8] x W(256,8)^T + b)
      long tiles = MT4 * (256 / 16);
      k_gemm<1, true, false, false, false, 4><<<(int)((tiles + 7) / 8), 256, 0, stream>>>(
          xdbl, dt_w, dt_b, dt, 256, 8, 40, 8, 256, 256 / 16, tiles,
          0, 0, 0, 0, 0);
    }
    k_scan<<<BB, 256, 0, stream>>>(u, dt, xdbl, xz, A_log, Dp, sy);
    { // out_proj: (BL,128) = sy(BL,256) x W(128,256)^T
      long tiles = MT4 * (128 / 16);
      k_gemm<0, false, false, false, false, 4><<<(int)((tiles + 7) / 8), 256, 0, stream>>>(
          sy, out_proj, nullptr, ym, 128, 256, 256, 256, 128, 128 / 16, tiles,
          0, 0, 0, 0, 0);
    }
    k_ln<<<BLM, 128, 0, stream>>>(tin, ym, ln1g, ln1b, ln2g, ln2b, tfCp);
    // te grouped conv as WMMA: groups=4, K=96 (3 blocks of 32 @ stride 128)
    k_wpack_te<<<(128 * 96 + 255) / 256, 256, 0, stream>>>(te_w, wpte);
    { // conv raw -> sy (scan output no longer needed)
      long tiles = MT4 * (128 / 16);
      k_gemm<0, true, true, false, true, 4><<<(int)((tiles + 7) / 8), 256, 0, stream>>>(
          tfCp, wpte, te_b, sy, 128, 96, 128, 96, 128, 128 / 16, tiles,
          LL, PSTR, 128, /*gct=*/2, /*gcs=*/32);
    }
    k_te_epi<<<(BLM * DM) / 256, 256, 0, stream>>>(sy, tfCp, te_bng, te_bnb, tout);

    float* t = tin; tin = tout; tout = t;
  }
  // final tf in `tin`, layout (B, L, 128)

  k_ca_attn<<<BB, 256, 0, stream>>>(tin, ca_w1, ca_w2, a_attn);
  k_padzero<<<BB, 256, 0, stream>>>(attp);
  k_ca_enh<<<(BLM * DM) / 256, 256, 0, stream>>>(tin, a_attn, ca_enhw, ca_enhb,
                                                 ca_bng, ca_bnb, attp);
  k_wpack_ae<<<(128 * 384 + 255) / 256, 256, 0, stream>>>(ae_w, wpae);
  { // ae conv as implicit-im2col GEMM: (BL,128) = attp-windows(BL,384) x Wp^T
    long tiles = MT4 * (128 / 16);
    k_gemm<0, true, true, false, false, 4><<<(int)((tiles + 7) / 8), 256, 0, stream>>>(
        attp, wpae, ae_b, convraw, 128, 384, 128, 384, 128, 128 / 16, tiles,
        LL, PSTR, 0, 0, 0);
  }
  k_ae_epi<<<(BB * DM * LL) / 256, 256, 0, stream>>>(convraw, attp, ae_bng, ae_bnb, enh);
  k_pool<<<BB * DM, 256, 0, stream>>>(enh, pooled, gf);
}